// DecoderSplatting_59416577573115
// MI455X (gfx1250) — compile-verified
//
#include <hip/hip_runtime.h>
#include <cstdint>
#include <cstddef>

#define VIEWS 8
#define IMG_H 480
#define IMG_W 640
#define HW    (IMG_H * IMG_W)      // 307200
#define NCH   14
#define OCH   15
#define TPB   256
#define PXT   4                    // pixels per thread
#define TILE  (TPB * PXT)          // 1024 pixels per block
#define TILES_PER_VIEW (HW / TILE) // 300, exact

typedef float v4f __attribute__((ext_vector_type(4)));

__device__ __forceinline__ float fsigmoid(float x) {
  return __builtin_amdgcn_rcpf(1.0f + __expf(-x));
}
__device__ __forceinline__ float fsoftplus(float x) {
  // numerically stable: max(x,0) + log1p(exp(-|x|))
  return fmaxf(x, 0.0f) + __logf(1.0f + __expf(-fabsf(x)));
}

__global__ __launch_bounds__(TPB) void decoder_splat_kernel(
    const float* __restrict__ raw,   // (1, V, 14, H, W)
    const float* __restrict__ ext,   // (1, V, 4, 4)
    const float* __restrict__ intr,  // (1, V, 3, 3)
    float* __restrict__ out)         // (V, H, W, 15)
{
  __shared__ v4f   sin[NCH * TPB];   // staged channel data: [c][thread] -> 4 pixels
  __shared__ float sc[32];           // per-view constants

  const int tid   = threadIdx.x;
  const int tile  = blockIdx.x;
  const int v     = tile / TILES_PER_VIEW;
  const int pbase = (tile - v * TILES_PER_VIEW) * TILE;

  // ---------- CDNA5 async global->LDS staging of 14 channel planes ----------
  {
    const float* gsrc = raw + (size_t)v * NCH * HW + (size_t)pbase + (size_t)tid * PXT;
    const uint32_t lbase = (uint32_t)(uintptr_t)(&sin[0]);
#pragma unroll
    for (int c = 0; c < NCH; ++c) {
      uint64_t ga = (uint64_t)(uintptr_t)(gsrc + (size_t)c * HW);
      uint32_t la = lbase + (uint32_t)((c * TPB + tid) * 16);
      asm volatile("global_load_async_to_lds_b128 %0, %1, off"
                   :: "v"(la), "v"(ga) : "memory");
    }
  }

  // ---------- per-view constants (one thread per block) ----------
  if (tid == 0) {
    const float* E = ext + v * 16;
#pragma unroll
    for (int k = 0; k < 16; ++k) sc[k] = E[k];
    const float* I = intr + v * 9;
    float a = I[0], b = I[1], c = I[2];
    float d = I[3], e = I[4], f = I[5];
    float g = I[6], h = I[7], i = I[8];
    float det = a * (e * i - f * h) - b * (d * i - f * g) + c * (d * h - e * g);
    float rd = 1.0f / det;
    sc[16] = (e * i - f * h) * rd; sc[17] = (c * h - b * i) * rd; sc[18] = (b * f - c * e) * rd;
    sc[19] = (f * g - d * i) * rd; sc[20] = (a * i - c * g) * rd; sc[21] = (c * d - a * f) * rd;
    sc[22] = (d * h - e * g) * rd; sc[23] = (b * g - a * h) * rd; sc[24] = (a * e - b * d) * rd;
    float det2 = a * e - b * d;                 // inv of 2x2 [[a,b],[d,e]], summed
    sc[25] = (a + e - b - d) / det2;
  }

  // own-wave staged data: wait ASYNCcnt; constants: workgroup barrier
  asm volatile("s_wait_asynccnt 0x0" ::: "memory");
  __syncthreads();

  const float e00 = sc[0],  e01 = sc[1],  e02 = sc[2],  e03 = sc[3];
  const float e10 = sc[4],  e11 = sc[5],  e12 = sc[6],  e13 = sc[7];
  const float e20 = sc[8],  e21 = sc[9],  e22 = sc[10], e23 = sc[11];
  const float e30 = sc[12], e31 = sc[13], e32 = sc[14], e33 = sc[15];
  const float ki00 = sc[16], ki01 = sc[17], ki02 = sc[18];
  const float ki10 = sc[19], ki11 = sc[20], ki12 = sc[21];
  const float ki20 = sc[22], ki21 = sc[23], ki22 = sc[24];
  const float mult = sc[25];

  const float* sf = (const float*)sin;
  const int px0 = pbase + tid * PXT;

  float o[PXT * OCH];

#pragma unroll
  for (int p = 0; p < PXT; ++p) {
    const int pix = px0 + p;
    const float x = (float)(pix % IMG_W);
    const float y = (float)(pix / IMG_W);

    float ch[NCH];
#pragma unroll
    for (int c = 0; c < NCH; ++c)
      ch[c] = sf[(c * TPB + tid) * 4 + p];

    // ---- ray direction through inv(K), normalized, rotated to world ----
    const float cx = x + fsigmoid(ch[12]) - 0.5f;
    const float cy = y + fsigmoid(ch[13]) - 0.5f;
    float dx = ki00 * cx + ki01 * cy + ki02;
    float dy = ki10 * cx + ki11 * cy + ki12;
    float dz = ki20 * cx + ki21 * cy + ki22;
    const float invd = __builtin_amdgcn_rsqf(dx * dx + dy * dy + dz * dz);
    dx *= invd; dy *= invd; dz *= invd;
    const float wd0 = e00 * dx + e01 * dy + e02 * dz;
    const float wd1 = e10 * dx + e11 * dy + e12 * dz;
    const float wd2 = e20 * dx + e21 * dy + e22 * dz;
    const float wd3 = e30 * dx + e31 * dy + e32 * dz;

    // ---- depth from disparity: 1/(sig(d)*(1/NEAR-1/FAR)+1/FAR) ----
    const float depth = __builtin_amdgcn_rcpf(fsigmoid(ch[3]) * 19.95f + 0.05f);

    const int ob = p * OCH;
    o[ob + 0] = e03 + wd0 * depth;
    o[ob + 1] = e13 + wd1 * depth;
    o[ob + 2] = e23 + wd2 * depth;
    o[ob + 3] = e33 + wd3 * depth;      // homogeneous component kept by reference

    o[ob + 4] = fsoftplus(ch[0]);
    o[ob + 5] = fsoftplus(ch[1]);
    o[ob + 6] = fsoftplus(ch[2]);
    o[ob + 7] = fsigmoid(ch[4]);
    o[ob + 8]  = fsoftplus(ch[5]) * mult;
    o[ob + 9]  = fsoftplus(ch[6]) * mult;
    o[ob + 10] = fsoftplus(ch[7]) * mult;

    // ---- quaternion: normalize(+eps), quat->mat, rotate by ext R, mat->quat ----
    float qx = ch[8], qy = ch[9], qz = ch[10], qw = ch[11];
    {
      const float n1 = __builtin_amdgcn_sqrtf(qx * qx + qy * qy + qz * qz + qw * qw);
      const float r1 = __builtin_amdgcn_rcpf(n1 + 1e-8f);
      qx *= r1; qy *= r1; qz *= r1; qw *= r1;
      const float r2 = __builtin_amdgcn_rsqf(qx * qx + qy * qy + qz * qz + qw * qw);
      qx *= r2; qy *= r2; qz *= r2; qw *= r2;
    }
    const float c00 = 1.0f - 2.0f * (qy * qy + qz * qz);
    const float c01 = 2.0f * (qx * qy - qz * qw);
    const float c02 = 2.0f * (qx * qz + qy * qw);
    const float c10 = 2.0f * (qx * qy + qz * qw);
    const float c11 = 1.0f - 2.0f * (qx * qx + qz * qz);
    const float c12 = 2.0f * (qy * qz - qx * qw);
    const float c20 = 2.0f * (qx * qz - qy * qw);
    const float c21 = 2.0f * (qy * qz + qx * qw);
    const float c22 = 1.0f - 2.0f * (qx * qx + qy * qy);

    const float m00 = e00 * c00 + e01 * c10 + e02 * c20;
    const float m01 = e00 * c01 + e01 * c11 + e02 * c21;
    const float m02 = e00 * c02 + e01 * c12 + e02 * c22;
    const float m10 = e10 * c00 + e11 * c10 + e12 * c20;
    const float m11 = e10 * c01 + e11 * c11 + e12 * c21;
    const float m12 = e10 * c02 + e11 * c12 + e12 * c22;
    const float m20 = e20 * c00 + e21 * c10 + e22 * c20;
    const float m21 = e20 * c01 + e21 * c11 + e22 * c21;
    const float m22 = e20 * c02 + e21 * c12 + e22 * c22;

    const float tr = m00 + m11 + m22;
    // argmax([m00,m11,m22,tr]) with first-max tie-break
    float w0 = m21 - m12, x0 = 1.0f + m00 - m11 - m22, y0 = m01 + m10, z0 = m02 + m20;
    float best = m00;
    if (m11 > best) { best = m11; w0 = m02 - m20; x0 = m01 + m10; y0 = 1.0f + m11 - m00 - m22; z0 = m12 + m21; }
    if (m22 > best) { best = m22; w0 = m10 - m01; x0 = m02 + m20; y0 = m12 + m21; z0 = 1.0f + m22 - m00 - m11; }
    if (tr  > best) {             w0 = 1.0f + tr; x0 = m21 - m12; y0 = m02 - m20; z0 = m10 - m01; }
    const float rq = __builtin_amdgcn_rsqf(w0 * w0 + x0 * x0 + y0 * y0 + z0 * z0);
    o[ob + 11] = w0 * rq;
    o[ob + 12] = x0 * rq;
    o[ob + 13] = y0 * rq;
    o[ob + 14] = z0 * rq;
  }

  // ---- 4 pixels * 15 ch = 240 contiguous, 16B-aligned bytes: 15 NT b128 stores ----
  v4f* op = (v4f*)(out + (size_t)(v * HW + px0) * OCH);
#pragma unroll
  for (int k = 0; k < OCH; ++k) {
    v4f s4 = { o[k * 4 + 0], o[k * 4 + 1], o[k * 4 + 2], o[k * 4 + 3] };
    __builtin_nontemporal_store(s4, op + k);
  }
}

extern "C" void kernel_launch(void* const* d_in, const int* in_sizes, int n_in,
                              void* d_out, int out_size, void* d_ws, size_t ws_size,
                              hipStream_t stream) {
  (void)in_sizes; (void)n_in; (void)d_ws; (void)ws_size; (void)out_size;
  const float* raw  = (const float*)d_in[0];
  const float* ext  = (const float*)d_in[1];
  const float* intr = (const float*)d_in[2];
  float* out = (float*)d_out;
  dim3 grid(VIEWS * TILES_PER_VIEW);   // 2400 blocks, no tail
  decoder_splat_kernel<<<grid, TPB, 0, stream>>>(raw, ext, intr, out);
}